// AreaEmbedding_69793218560006
// MI455X (gfx1250) — compile-verified
//
#include <hip/hip_runtime.h>
#include <hip/hip_bf16.h>

// ---------------------------------------------------------------------------
// AreaEmbedding loss on MI455X (gfx1250, wave32).
//
//   loss = sum (wid-ken)^2 + (wid-lrg)^2 + (lrg-sml)^2 + (sml-yad)^2
//        + sum_{i, j in pos(i), k in neg(i)} relu(D2[i,j] - D2[i,k] + alpha)
//
// D2 comes from the Gram matrix Y*Y^T (512x512x128 f32 GEMM) which we run on
// the CDNA5 f32 matrix pipe via V_WMMA_F32_16X16X4_F32 (exact f32 semantics,
// matching the reference; the loss subtracts distances so cancellation-safe
// f32 accumulation matters more than raw matrix throughput here).
// ---------------------------------------------------------------------------

#define N_YAD 512
#define D_EMB 128
#define K_CO  16
#define ALPHA 0.1f

typedef __attribute__((ext_vector_type(2))) float v2f;
typedef __attribute__((ext_vector_type(8))) float v8f;

// ---------------------------------------------------------------- kernel 0
__global__ void ae_zero_out(float* out) {
    if (threadIdx.x == 0 && blockIdx.x == 0) out[0] = 0.0f;
}

// ---------------------------------------------------------------- kernel 1
// sq[i] = ||Y[i]||^2  (512 rows x 128). One thread per row.
__global__ void ae_rowsq(const float* __restrict__ Y, float* __restrict__ sq) {
    int row = blockIdx.x * blockDim.x + threadIdx.x;
    if (row >= N_YAD) return;
    const float* __restrict__ p = Y + row * D_EMB;
    float s = 0.0f;
#pragma unroll 8
    for (int d = 0; d < D_EMB; ++d) {
        float v = p[d];
        s = fmaf(v, v, s);
    }
    sq[row] = s;
}

// ---------------------------------------------------------------- kernel 2
// Hierarchical consistency term, block-reduced then atomically accumulated.
__global__ void ae_hier(const float* __restrict__ wid,
                        const float* __restrict__ ken,
                        const float* __restrict__ lrg,
                        const float* __restrict__ sml,
                        const float* __restrict__ yad,
                        float* __restrict__ out, int n) {
    __shared__ float red[256];
    int idx = blockIdx.x * blockDim.x + threadIdx.x;
    int stride = gridDim.x * blockDim.x;
    float s = 0.0f;
    for (int i = idx; i < n; i += stride) {
        float w = wid[i], k = ken[i], l = lrg[i], m = sml[i], y = yad[i];
        float d1 = w - k, d2 = w - l, d3 = l - m, d4 = m - y;
        s = fmaf(d1, d1, s);
        s = fmaf(d2, d2, s);
        s = fmaf(d3, d3, s);
        s = fmaf(d4, d4, s);
    }
    red[threadIdx.x] = s;
    __syncthreads();
    for (int off = 128; off > 0; off >>= 1) {
        if (threadIdx.x < (unsigned)off) red[threadIdx.x] += red[threadIdx.x + off];
        __syncthreads();
    }
    if (threadIdx.x == 0) atomicAdd(out, red[0]);
}

// ---------------------------------------------------------------- kernel 3
// D2 = sq_i + sq_j - 2 * (Y Y^T) via V_WMMA_F32_16X16X4_F32.
// One wave (32 lanes) per 16x16 output tile; K-loop of 32 wmma steps.
//
// f32 A-matrix 16x4 layout (ISA 7.12.2): lanes 0-15 hold M=0..15 with
// VGPR0=K0,VGPR1=K1; lanes 16-31 hold K2/K3.  B (4x16) is symmetric with
// rows of the j-tile, so both fragments are "row m = lane&15, k-pair
// selected by lane>>4" loads — a single v2f (b64) load each.
__global__ __launch_bounds__(32)
void ae_gram_wmma(const float* __restrict__ Y,
                  const float* __restrict__ sq,
                  float* __restrict__ D2) {
    const int tileJ = blockIdx.x;        // 0..31
    const int tileI = blockIdx.y;        // 0..31
    const int lane  = threadIdx.x;       // 0..31 (one wave)
    const int m     = lane & 15;
    const int half  = lane >> 4;         // selects K pair {0,1} vs {2,3}

    const float* __restrict__ arow = Y + (tileI * 16 + m) * D_EMB + 2 * half;
    const float* __restrict__ brow = Y + (tileJ * 16 + m) * D_EMB + 2 * half;

    v8f acc = {};
#pragma unroll 4
    for (int k = 0; k < D_EMB; k += 4) {
        v2f a = *(const v2f*)(arow + k);
        v2f b = *(const v2f*)(brow + k);
        // (neg_a, A, neg_b, B, c_mod, C, reuse_a, reuse_b)
        acc = __builtin_amdgcn_wmma_f32_16x16x4_f32(
            false, a, false, b, (short)0, acc, false, false);
    }

    // C/D layout: acc[r] = D[M = r + 8*half][Nc = lane&15]
    const int gj = tileJ * 16 + m;
    const float sqj = sq[gj];
#pragma unroll
    for (int r = 0; r < 8; ++r) {
        const int gi = tileI * 16 + r + 8 * half;
        D2[gi * N_YAD + gj] = sq[gi] + sqj - 2.0f * acc[r];
    }
}

// ---------------------------------------------------------------- kernel 4
// Triplet hinge: one block per anchor i. LDS holds the D2 row, a 512-bit
// membership bitmask (dedups x[i]), and the <=16 positive distances.
__global__ __launch_bounds__(256)
void ae_triplet(const float* __restrict__ D2,
                const int* __restrict__ x,
                float* __restrict__ out) {
    __shared__ float    row[N_YAD];
    __shared__ unsigned mask[N_YAD / 32];   // 512-bit membership bitmask
    __shared__ float    posd[K_CO];
    __shared__ int      posn;
    __shared__ float    red[256];

    const int i   = blockIdx.x;
    const int tid = threadIdx.x;

    for (int j = tid; j < N_YAD; j += blockDim.x) row[j] = D2[i * N_YAD + j];
    if (tid < N_YAD / 32) mask[tid] = 0u;
    __syncthreads();

    if (tid < K_CO) {
        int j = x[i * K_CO + tid] & (N_YAD - 1);
        atomicOr(&mask[j >> 5], 1u << (j & 31));
    }
    __syncthreads();

    if (tid == 0) {
        int cnt = 0;
        for (int w = 0; w < N_YAD / 32; ++w) {
            unsigned mw = mask[w];
            while (mw) {
                int b = __ffs(mw) - 1;
                mw &= mw - 1;
                int j = w * 32 + b;
                if (j != i) posd[cnt++] = row[j];   // pos = member & (j != i)
            }
        }
        posn = cnt;
    }
    __syncthreads();

    const int P = posn;
    float s = 0.0f;
    for (int k = tid; k < N_YAD; k += blockDim.x) {
        if (k == i) continue;                              // ~eye
        if (mask[k >> 5] & (1u << (k & 31))) continue;     // neg = ~member
        const float dik = row[k];
#pragma unroll
        for (int p = 0; p < K_CO; ++p) {
            if (p >= P) break;
            float t = posd[p] - dik + ALPHA;
            s += (t > 0.0f) ? t : 0.0f;
        }
    }

    red[tid] = s;
    __syncthreads();
    for (int off = 128; off > 0; off >>= 1) {
        if (tid < (unsigned)off) red[tid] += red[tid + off];
        __syncthreads();
    }
    if (tid == 0) atomicAdd(out, red[0]);
}

// ---------------------------------------------------------------------------
extern "C" void kernel_launch(void* const* d_in, const int* in_sizes, int n_in,
                              void* d_out, int out_size, void* d_ws, size_t ws_size,
                              hipStream_t stream) {
    const float* wid = (const float*)d_in[0];
    const float* ken = (const float*)d_in[1];
    const float* lrg = (const float*)d_in[2];
    const float* sml = (const float*)d_in[3];
    const float* yad = (const float*)d_in[4];
    const int*   x   = (const int*)d_in[5];
    float* out = (float*)d_out;

    // Workspace: [ sq : 512 f32 | D2 : 512*512 f32 ]  (~1.05 MB)
    float* sq = (float*)d_ws;
    float* D2 = sq + N_YAD;

    ae_zero_out<<<1, 32, 0, stream>>>(out);

    ae_rowsq<<<(N_YAD + 255) / 256, 256, 0, stream>>>(yad, sq);

    ae_hier<<<64, 256, 0, stream>>>(wid, ken, lrg, sml, yad, out, N_YAD * D_EMB);

    dim3 ggrid(N_YAD / 16, N_YAD / 16);   // 32 x 32 tiles, one wave each
    ae_gram_wmma<<<ggrid, 32, 0, stream>>>(yad, sq, D2);

    ae_triplet<<<N_YAD, 256, 0, stream>>>(D2, x, out);
}